// FALoss1_86706799772428
// MI455X (gfx1250) — compile-verified
//
#include <hip/hip_runtime.h>

typedef float v2f __attribute__((ext_vector_type(2)));
typedef float v8f __attribute__((ext_vector_type(8)));

#define HDIM 512
#define WDIM 512
#define CHANNELS 256
#define BATCH 8
#define NBLK 1024          // 32x32 pooled blocks per plane
#define WAVES_PER_WG 8
#define NPART 64           // partials per channel = BATCH * WAVES_PER_WG
#define NPOOL 8192.0f      // pooled elements per channel = 8 * 32 * 32

// Sum a 16x16 block of each feature using V_WMMA_F32_16X16X4_F32.
// A = all-ones 16x4, so D[m][n] = column-sum of B regardless of which block
// element lands in which B slot -> we may use any bijection of the 256 block
// elements onto the 4 WMMAs' B operands. We pick the one produced by fully
// coalesced float4 loads (lane L reads row L/4, cols 4*(L%4), and row+8).
__device__ __forceinline__ void block_sums2(const float* __restrict__ p1,
                                            const float* __restrict__ p2,
                                            int row0, int col0,
                                            float& out1, float& out2) {
    const int lane = threadIdx.x & 31;
    const int r    = lane >> 2;          // 0..7
    const int cofs = (lane & 3) << 2;    // 0,4,8,12
    const size_t off = (size_t)(row0 + r) * WDIM + (size_t)(col0 + cofs);

    // 4 x global_load_b128: stream both features' halves of the block
    const float4 a0 = *(const float4*)(p1 + off);
    const float4 a1 = *(const float4*)(p1 + off + 8 * WDIM);
    const float4 b0 = *(const float4*)(p2 + off);
    const float4 b1 = *(const float4*)(p2 + off + 8 * WDIM);

    v2f ones; ones.x = 1.0f; ones.y = 1.0f;
    v8f c1 = {};
    v8f c2 = {};
    v2f t;

    t.x = a0.x; t.y = a0.y;
    c1 = __builtin_amdgcn_wmma_f32_16x16x4_f32(false, ones, false, t, (short)0, c1, false, false);
    t.x = a0.z; t.y = a0.w;
    c1 = __builtin_amdgcn_wmma_f32_16x16x4_f32(false, ones, false, t, (short)0, c1, false, false);
    t.x = a1.x; t.y = a1.y;
    c1 = __builtin_amdgcn_wmma_f32_16x16x4_f32(false, ones, false, t, (short)0, c1, false, false);
    t.x = a1.z; t.y = a1.w;
    c1 = __builtin_amdgcn_wmma_f32_16x16x4_f32(false, ones, false, t, (short)0, c1, false, false);

    t.x = b0.x; t.y = b0.y;
    c2 = __builtin_amdgcn_wmma_f32_16x16x4_f32(false, ones, false, t, (short)0, c2, false, false);
    t.x = b0.z; t.y = b0.w;
    c2 = __builtin_amdgcn_wmma_f32_16x16x4_f32(false, ones, false, t, (short)0, c2, false, false);
    t.x = b1.x; t.y = b1.y;
    c2 = __builtin_amdgcn_wmma_f32_16x16x4_f32(false, ones, false, t, (short)0, c2, false, false);
    t.x = b1.z; t.y = b1.w;
    c2 = __builtin_amdgcn_wmma_f32_16x16x4_f32(false, ones, false, t, (short)0, c2, false, false);

    // c[0] in lane L = column-sum of column (L % 16); fold 16 columns.
    float s1 = c1[0];
    float s2 = c2[0];
    s1 += __shfl_xor(s1, 1, 32);  s2 += __shfl_xor(s2, 1, 32);
    s1 += __shfl_xor(s1, 2, 32);  s2 += __shfl_xor(s2, 2, 32);
    s1 += __shfl_xor(s1, 4, 32);  s2 += __shfl_xor(s2, 4, 32);
    s1 += __shfl_xor(s1, 8, 32);  s2 += __shfl_xor(s2, 8, 32);
    out1 = s1;
    out2 = s2;
}

// One workgroup per (b, c) plane; each of the 8 waves handles 128 of the 1024
// pooled blocks, accumulating the 5 moments in registers, then writes its
// partials to a fixed workspace slot (deterministic, no float atomics).
__global__ void __launch_bounds__(256)
fa_pool_stats_kernel(const float* __restrict__ f1,
                     const float* __restrict__ f2,
                     float* __restrict__ partials) {
    const int plane = blockIdx.x;        // b-major: plane = b*256 + c
    const int b = plane >> 8;
    const int ch = plane & 255;
    const int wave = threadIdx.x >> 5;   // 0..7

    const float* p1 = f1 + (size_t)plane * (HDIM * WDIM);
    const float* p2 = f2 + (size_t)plane * (HDIM * WDIM);

    float s1 = 0.f, s2 = 0.f, s11 = 0.f, s22 = 0.f, s12 = 0.f;

    for (int bi = wave; bi < NBLK; bi += WAVES_PER_WG) {
        const int row0 = (bi >> 5) << 4;
        const int col0 = (bi & 31) << 4;
        float t1, t2;
        block_sums2(p1, p2, row0, col0, t1, t2);
        const float a1 = t1 * (1.0f / 256.0f);  // pooled value
        const float a2 = t2 * (1.0f / 256.0f);
        s1 += a1;
        s2 += a2;
        s11 = fmaf(a1, a1, s11);
        s22 = fmaf(a2, a2, s22);
        s12 = fmaf(a1, a2, s12);
    }

    if ((threadIdx.x & 31) == 0) {
        const int pidx = ch * NPART + (b * WAVES_PER_WG + wave);  // [c][64]
        partials[0 * CHANNELS * NPART + pidx] = s1;
        partials[1 * CHANNELS * NPART + pidx] = s2;
        partials[2 * CHANNELS * NPART + pidx] = s11;
        partials[3 * CHANNELS * NPART + pidx] = s22;
        partials[4 * CHANNELS * NPART + pidx] = s12;
    }
}

// One thread per channel: deterministic reduction of the 64 partials,
// correlation, then LDS tree-reduction of |corr| to the scalar output.
__global__ void __launch_bounds__(256)
fa_finalize_kernel(const float* __restrict__ partials, float* __restrict__ out) {
    __shared__ float red[256];
    const int ch = threadIdx.x;

    float s1 = 0.f, s2 = 0.f, s11 = 0.f, s22 = 0.f, s12 = 0.f;
    const int base = ch * NPART;
    for (int i = 0; i < NPART; ++i) {
        s1  += partials[0 * CHANNELS * NPART + base + i];
        s2  += partials[1 * CHANNELS * NPART + base + i];
        s11 += partials[2 * CHANNELS * NPART + base + i];
        s22 += partials[3 * CHANNELS * NPART + base + i];
        s12 += partials[4 * CHANNELS * NPART + base + i];
    }

    const float n  = NPOOL;
    const float m1 = s1 / n;
    const float m2 = s2 / n;
    const float cov  = s12 / n - m1 * m2;                 // biased (torch.mean)
    const float var1 = (s11 - n * m1 * m1) / (n - 1.0f);  // unbiased (torch.std)
    const float var2 = (s22 - n * m2 * m2) / (n - 1.0f);
    const float corr = cov / (sqrtf(var1) * sqrtf(var2) + 1e-8f);

    red[ch] = fabsf(corr);
    __syncthreads();
    for (int offs = 128; offs > 0; offs >>= 1) {
        if (ch < offs) red[ch] += red[ch + offs];
        __syncthreads();
    }
    if (ch == 0) out[0] = 1.0f - red[0] * (1.0f / 256.0f);
}

extern "C" void kernel_launch(void* const* d_in, const int* in_sizes, int n_in,
                              void* d_out, int out_size, void* d_ws, size_t ws_size,
                              hipStream_t stream) {
    const float* f1 = (const float*)d_in[0];
    const float* f2 = (const float*)d_in[1];
    float* partials = (float*)d_ws;   // needs 5*256*64*4 = 320 KiB
    float* out = (float*)d_out;

    fa_pool_stats_kernel<<<BATCH * CHANNELS, 256, 0, stream>>>(f1, f2, partials);
    fa_finalize_kernel<<<1, 256, 0, stream>>>(partials, out);
}